// GraphConvolutionLayer_44384192037512
// MI455X (gfx1250) — compile-verified
//
#include <hip/hip_runtime.h>

#define NNODE 2048
#define NCH   512
#define NB    8
#define NM    (NB * NCH)   // 4096 rows of x / out
#define NNZE  65536
#define LDST  20           // LDS row stride in floats (conflict-free, 80B = 16B-aligned rows)

typedef __attribute__((ext_vector_type(2))) float v2f;
typedef __attribute__((ext_vector_type(8))) float v8f;

#if __has_builtin(__builtin_amdgcn_global_load_async_to_lds_b128) && \
    __has_builtin(__builtin_amdgcn_s_wait_asynccnt)
#define HAVE_ASYNC 1
#else
#define HAVE_ASYNC 0
#endif

// ---------------------------------------------------------------------------
// K1: row_ptr[r] = lower_bound(edge_idx, r*N)  (edge_idx sorted, unique)
// ---------------------------------------------------------------------------
__global__ void k_rowptr(const long long* __restrict__ idx, int* __restrict__ row_ptr) {
    int r = blockIdx.x * blockDim.x + threadIdx.x;
    if (r > NNODE) return;
    long long target = (long long)r * NNODE;
    int lo = 0, hi = NNZE;
    while (lo < hi) {
        int mid = (lo + hi) >> 1;
        if (idx[mid] < target) lo = mid + 1; else hi = mid;
    }
    row_ptr[r] = lo;
}

// ---------------------------------------------------------------------------
// K2: per-row softmax over that row's edges (masked entries underflow to 0).
// ---------------------------------------------------------------------------
__global__ void k_softmax(const long long* __restrict__ idx, const float* __restrict__ e,
                          const int* __restrict__ row_ptr, float* __restrict__ w,
                          float* __restrict__ diag) {
    int r = blockIdx.x;
    int s = row_ptr[r], t = row_ptr[r + 1];
    int lane = threadIdx.x;
    if (t == s) { if (lane == 0) diag[r] = 0.0f; return; }

    float mx = -3.0e38f;
    for (int j = s + lane; j < t; j += 32) mx = fmaxf(mx, e[j]);
    for (int o = 16; o; o >>= 1) mx = fmaxf(mx, __shfl_xor(mx, o, 32));

    float sum = 0.0f;
    for (int j = s + lane; j < t; j += 32) sum += expf(e[j] - mx);
    for (int o = 16; o; o >>= 1) sum += __shfl_xor(sum, o, 32);
    float inv = 1.0f / sum;

    float selfw = 0.0f;
    for (int j = s + lane; j < t; j += 32) {
        float wj = expf(e[j] - mx) * inv;
        w[j] = wj;
        int col = (int)(idx[j] - (long long)r * NNODE);
        if (col == r) selfw += wj;
    }
    for (int o = 16; o; o >>= 1) selfw += __shfl_xor(selfw, o, 32);
    if (lane == 0) diag[r] = selfw;
}

// ---------------------------------------------------------------------------
// K3: dst[n][k] = src[k][n] * (scale ? scale[n] : 1)   (LDS-tiled transpose)
// ---------------------------------------------------------------------------
__global__ __launch_bounds__(256) void k_transpose_scale(const float* __restrict__ src,
                                                         float* __restrict__ dst,
                                                         const float* __restrict__ scale) {
    __shared__ float tile[32][33];
    int n0 = blockIdx.x * 32;
    int k0 = blockIdx.y * 32;
    int tx = threadIdx.x, ty = threadIdx.y;   // 32 x 8
#pragma unroll
    for (int i = 0; i < 4; i++)
        tile[ty + i * 8][tx] = src[(long)(k0 + ty + i * 8) * NNODE + n0 + tx];
    __syncthreads();
#pragma unroll
    for (int i = 0; i < 4; i++) {
        int n = n0 + ty + i * 8;
        float sc = scale ? scale[n] : 1.0f;
        dst[(long)n * NNODE + k0 + tx] = tile[tx][ty + i * 8] * sc;
    }
}

// ---------------------------------------------------------------------------
// K4: WeffT[n][:] += w_e * W2T[m][:] for every off-diagonal edge (m,n).
// ---------------------------------------------------------------------------
__global__ __launch_bounds__(256) void k_scatter(const long long* __restrict__ idx,
                                                 const float* __restrict__ w,
                                                 const float* __restrict__ W2T,
                                                 float* __restrict__ WeffT) {
    int eidx = blockIdx.x;
    long long fi = idx[eidx];
    int m = (int)(fi >> 11);
    int n = (int)(fi & (NNODE - 1));
    if (m == n) return;                       // A_off has zero diagonal
    float wv = w[eidx];
    const float* srow = W2T + (long)m * NNODE;
    float* drow = WeffT + (long)n * NNODE;
    for (int k = threadIdx.x; k < NNODE; k += 256)
        atomicAdd(&drow[k], srow[k] * wv);
}

// ---------------------------------------------------------------------------
// K5: OUT = X @ Weff with V_WMMA_F32_16X16X4_F32.
// Block = 128 thr (4 waves) -> 128x128 tile; wave -> 64x64 (4x4 WMMA tiles).
// Double-buffered LDS; next chunk prefetched via GLOBAL_LOAD_ASYNC_TO_LDS_B128
// (ASYNCcnt pipeline).
// ---------------------------------------------------------------------------
__device__ __forceinline__ void wmma_chunk(const float* Abuf, const float* Bbuf,
                                           int waveM, int waveN, int half, int l,
                                           v8f (&acc)[4][4]) {
#pragma unroll
    for (int ks = 0; ks < 4; ks++) {
        const int k = ks * 4 + half * 2;
        v2f a[4], b[4];
#pragma unroll
        for (int t = 0; t < 4; t++)
            a[t] = *(const v2f*)(Abuf + (waveM + t * 16 + l) * LDST + k);
#pragma unroll
        for (int u = 0; u < 4; u++)
            b[u] = *(const v2f*)(Bbuf + (waveN + u * 16 + l) * LDST + k);
#pragma unroll
        for (int t = 0; t < 4; t++)
#pragma unroll
            for (int u = 0; u < 4; u++)
                acc[t][u] = __builtin_amdgcn_wmma_f32_16x16x4_f32(
                    false, a[t], false, b[u], (short)0, acc[t][u], false, false);
    }
}

#if HAVE_ASYNC
typedef int v4i __attribute__((__vector_size__(16)));
typedef __attribute__((address_space(1))) v4i* gv4i_p;   // global int4*
typedef __attribute__((address_space(3))) v4i* lv4i_p;   // LDS int4*

__device__ __forceinline__ void issue_tile(const float* X, const float* BT,
                                           int block_m, int block_n, int kc, int tid,
                                           float* Abuf, float* Bbuf) {
    // A/B tiles: 128 rows x 16 floats each = 512 float4 per matrix; 4 per thread.
#pragma unroll
    for (int it = 0; it < 4; it++) {
        int q = tid + it * 128;
        int row = q >> 2, kq = q & 3;
        __builtin_amdgcn_global_load_async_to_lds_b128(
            (gv4i_p)(X + (long)(block_m + row) * NNODE + kc + kq * 4),
            (lv4i_p)(Abuf + row * LDST + kq * 4), 0, 0);
        __builtin_amdgcn_global_load_async_to_lds_b128(
            (gv4i_p)(BT + (long)(block_n + row) * NNODE + kc + kq * 4),
            (lv4i_p)(Bbuf + row * LDST + kq * 4), 0, 0);
    }
}
#endif

__global__ __launch_bounds__(128) void k_gemm_wmma(const float* __restrict__ X,
                                                   const float* __restrict__ BT,
                                                   float* __restrict__ OUT) {
    __shared__ float As[2][128 * LDST];
    __shared__ float Bs[2][128 * LDST];

    const int tid  = threadIdx.x;
    const int wave = tid >> 5;
    const int lane = tid & 31;
    const int half = lane >> 4;
    const int l    = lane & 15;

    const int block_m = blockIdx.y * 128;
    const int block_n = blockIdx.x * 128;
    const int waveM = (wave & 1) * 64;    // 0 or 64
    const int waveN = (wave >> 1) * 64;   // 0 or 64

    v8f acc[4][4] = {};

#if HAVE_ASYNC
    // ---- ASYNCcnt-pipelined double buffering ----
    issue_tile(X, BT, block_m, block_n, 0, tid, As[0], Bs[0]);   // asynccnt: 8
#pragma unroll 1
    for (int kc = 0; kc < NNODE; kc += 32) {
        issue_tile(X, BT, block_m, block_n, (kc + 16) & (NNODE - 1), tid, As[1], Bs[1]);
        __builtin_amdgcn_s_wait_asynccnt(8);   // chunk kc landed (in-order completion)
        __syncthreads();
        wmma_chunk(As[0], Bs[0], waveM, waveN, half, l, acc);
        __syncthreads();

        issue_tile(X, BT, block_m, block_n, (kc + 32) & (NNODE - 1), tid, As[0], Bs[0]);
        __builtin_amdgcn_s_wait_asynccnt(8);   // chunk kc+16 landed
        __syncthreads();
        wmma_chunk(As[1], Bs[1], waveM, waveN, half, l, acc);
        __syncthreads();
    }
#else
    // ---- fallback: register prefetch + single LDS buffer ----
    float4 ra[4], rb[4];
#pragma unroll
    for (int it = 0; it < 4; it++) {
        int q = tid + it * 128;
        int row = q >> 2, kq = q & 3;
        ra[it] = *(const float4*)(X  + (long)(block_m + row) * NNODE + kq * 4);
        rb[it] = *(const float4*)(BT + (long)(block_n + row) * NNODE + kq * 4);
    }
#pragma unroll 1
    for (int kc = 0; kc < NNODE; kc += 16) {
        __syncthreads();
#pragma unroll
        for (int it = 0; it < 4; it++) {
            int q = tid + it * 128;
            int row = q >> 2, kq = q & 3;
            *(float4*)(As[0] + row * LDST + kq * 4) = ra[it];
            *(float4*)(Bs[0] + row * LDST + kq * 4) = rb[it];
        }
        __syncthreads();
        if (kc + 16 < NNODE) {
#pragma unroll
            for (int it = 0; it < 4; it++) {
                int q = tid + it * 128;
                int row = q >> 2, kq = q & 3;
                ra[it] = *(const float4*)(X  + (long)(block_m + row) * NNODE + kc + 16 + kq * 4);
                rb[it] = *(const float4*)(BT + (long)(block_n + row) * NNODE + kc + 16 + kq * 4);
            }
        }
        wmma_chunk(As[0], Bs[0], waveM, waveN, half, l, acc);
    }
#endif

    // C/D layout: VGPR v, lanes 0-15 -> (M=v, N=l); lanes 16-31 -> (M=v+8, N=l)
#pragma unroll
    for (int t = 0; t < 4; t++)
#pragma unroll
        for (int u = 0; u < 4; u++)
#pragma unroll
            for (int v = 0; v < 8; v++) {
                int row = block_m + waveM + t * 16 + v + 8 * half;
                int col = block_n + waveN + u * 16 + l;
                OUT[(long)row * NNODE + col] = acc[t][u][v];
            }
}

// ---------------------------------------------------------------------------
extern "C" void kernel_launch(void* const* d_in, const int* in_sizes, int n_in,
                              void* d_out, int out_size, void* d_ws, size_t ws_size,
                              hipStream_t stream) {
    const float*     x        = (const float*)d_in[0];
    const float*     W1       = (const float*)d_in[1];
    const float*     W2       = (const float*)d_in[2];
    const float*     e        = (const float*)d_in[3];
    const long long* edge_idx = (const long long*)d_in[4];
    float*           out      = (float*)d_out;

    char* ws = (char*)d_ws;
    int*   row_ptr = (int*)  (ws + 0);                       // 2049 ints
    float* diag    = (float*)(ws + 16384);                   // 2048 f32
    float* w       = (float*)(ws + 32768);                   // 65536 f32
    float* WeffT   = (float*)(ws + 524288);                  // 16 MB
    float* W2T     = (float*)(ws + 524288 + 16777216);       // 16 MB

    k_rowptr<<<(NNODE + 1 + 255) / 256, 256, 0, stream>>>(edge_idx, row_ptr);
    k_softmax<<<NNODE, 32, 0, stream>>>(edge_idx, e, row_ptr, w, diag);

    dim3 tgrid(NNODE / 32, NNODE / 32);
    k_transpose_scale<<<tgrid, dim3(32, 8), 0, stream>>>(W1, WeffT, diag);    // WeffT = W1^T * diag
    k_transpose_scale<<<tgrid, dim3(32, 8), 0, stream>>>(W2, W2T, nullptr);   // W2T = W2^T

    k_scatter<<<NNZE, 256, 0, stream>>>(edge_idx, w, W2T, WeffT);             // += W2 @ A_off (transposed)

    k_gemm_wmma<<<dim3(NNODE / 128, NM / 128), 128, 0, stream>>>(x, WeffT, out);
}